// GraphAttn_21706764714530
// MI455X (gfx1250) — compile-verified
//
#include <hip/hip_runtime.h>
#include <hip/hip_bf16.h>

// GAT forward for MI455X (gfx1250, wave32).
//   h1 = x @ W^T ; h2 = n @ W^T           (f32 WMMA 16x16x4)
//   s1 = h1 @ w_att1 ; s2 = h2 @ w_att2   (in-register shfl reduce; h2 never stored)
//   out[i,:] = softmax_j(LeakyReLU(s1_i + s2_j)) @ h1
// Softmax row-max uses m_i = f(s1_i + max_j s2_j) (LeakyReLU is monotone), so all
// exponents are <= 0 and only one global max over s2[8192] is needed.

#define N_NODES 8192
#define IN_F    512
#define OUT_F   64
#define ALPHA   0.2f
#define JCHUNK  128   // keys staged in LDS per block iteration (128*64*4 = 32 KB)

typedef __attribute__((ext_vector_type(2))) float v2f;
typedef __attribute__((ext_vector_type(4))) float v4f;
typedef __attribute__((ext_vector_type(8))) float v8f;

// D(16x16,f32) = A(16x4,f32) * B(4x16,f32) + C   — full-precision matrix op.
__device__ __forceinline__ v8f wmma4f32(v2f a, v2f b, v8f c) {
    return __builtin_amdgcn_wmma_f32_16x16x4_f32(
        /*neg_a=*/false, a, /*neg_b=*/false, b,
        /*c_mod=*/(short)0, c, /*reuse_a=*/false, /*reuse_b=*/false);
}

__device__ __forceinline__ float lrelu(float v) {
    return fmaxf(v, ALPHA * v);   // for v<0: 0.2v > v ; for v>=0: v >= 0.2v
}

// ---------------------------------------------------------------------------
// Kernel 1: projection GEMMs + attention score vectors.
// One wave computes a 16-row tile of h1 and h2 (16 x 64 each) with f32 WMMA.
// 256 threads = 8 waves per block -> block covers 128 rows; grid = 64 blocks.
// ---------------------------------------------------------------------------
__global__ __launch_bounds__(256) void k_proj(
    const float* __restrict__ x, const float* __restrict__ n,
    const float* __restrict__ W,                 // [OUT_F, IN_F] row-major
    const float* __restrict__ wa1, const float* __restrict__ wa2,
    float* __restrict__ h1, float* __restrict__ s1, float* __restrict__ s2)
{
    const int lane = threadIdx.x & 31;
    const int wave = threadIdx.x >> 5;
    const int i0   = (blockIdx.x * 8 + wave) * 16;
    const int lm   = lane & 15;        // M index within tile (both halves)
    const int half = lane >> 4;        // 0: K=0..1 side, 1: K=2..3 side
    const int ko   = half * 2;

    v8f c1[4] = {}, c2[4] = {};        // 4 feature tiles x {h1,h2}

    const float* xr = x + (size_t)(i0 + lm) * IN_F + ko;
    const float* nr = n + (size_t)(i0 + lm) * IN_F + ko;
    const float* wr = W + (size_t)lm * IN_F + ko;   // row (16t+lm) adds 16*t*IN_F

    for (int k0 = 0; k0 < IN_F; k0 += 4) {
        // A fragments (ISA 16x4 layout): lane<16 holds K=k0..k0+1, lane>=16 K=k0+2..k0+3.
        v2f a1 = *(const v2f*)(xr + k0);
        v2f a2 = *(const v2f*)(nr + k0);
        // B fragments mirror A: V0 = K=(k0+ko), V1 = K=(k0+ko+1); N = lane&15.
        v2f b0 = *(const v2f*)(wr + k0);
        v2f b1 = *(const v2f*)(wr + 16 * IN_F + k0);
        v2f b2 = *(const v2f*)(wr + 32 * IN_F + k0);
        v2f b3 = *(const v2f*)(wr + 48 * IN_F + k0);
        c1[0] = wmma4f32(a1, b0, c1[0]);  c2[0] = wmma4f32(a2, b0, c2[0]);
        c1[1] = wmma4f32(a1, b1, c1[1]);  c2[1] = wmma4f32(a2, b1, c2[1]);
        c1[2] = wmma4f32(a1, b2, c1[2]);  c2[2] = wmma4f32(a2, b2, c2[2]);
        c1[3] = wmma4f32(a1, b3, c1[3]);  c2[3] = wmma4f32(a2, b3, c2[3]);
    }

    // Store h1 tile. C layout: VGPR r -> row (r + half*8), col = 16t + lm.
    #pragma unroll
    for (int t = 0; t < 4; ++t)
        #pragma unroll
        for (int r = 0; r < 8; ++r)
            h1[(size_t)(i0 + r + half * 8) * OUT_F + 16 * t + lm] = c1[t][r];

    // s1/s2: dot each row with w_att vectors; reduce across the 16-lane half.
    float wa1v[4], wa2v[4];
    #pragma unroll
    for (int t = 0; t < 4; ++t) { wa1v[t] = wa1[16 * t + lm]; wa2v[t] = wa2[16 * t + lm]; }

    #pragma unroll
    for (int r = 0; r < 8; ++r) {
        float v1 = 0.f, v2 = 0.f;
        #pragma unroll
        for (int t = 0; t < 4; ++t) { v1 += c1[t][r] * wa1v[t]; v2 += c2[t][r] * wa2v[t]; }
        #pragma unroll
        for (int m = 1; m <= 8; m <<= 1) {   // xor masks 1,2,4,8: stay inside half
            v1 += __shfl_xor(v1, m, 32);
            v2 += __shfl_xor(v2, m, 32);
        }
        if (lm == 0) {
            s1[i0 + r + half * 8] = v1;
            s2[i0 + r + half * 8] = v2;
        }
    }
}

// ---------------------------------------------------------------------------
// Kernel 2: global max of s2 (single block, deterministic tree reduction).
// ---------------------------------------------------------------------------
__global__ __launch_bounds__(256) void k_s2max(const float* __restrict__ s2,
                                               float* __restrict__ s2max)
{
    __shared__ float red[256];
    float m = -3.402823466e+38f;
    for (int i = threadIdx.x; i < N_NODES; i += 256) m = fmaxf(m, s2[i]);
    red[threadIdx.x] = m;
    __syncthreads();
    for (int s = 128; s > 0; s >>= 1) {
        if (threadIdx.x < s) red[threadIdx.x] = fmaxf(red[threadIdx.x], red[threadIdx.x + s]);
        __syncthreads();
    }
    if (threadIdx.x == 0) s2max[0] = red[0];
}

// ---------------------------------------------------------------------------
// Kernel 3: streaming softmax-attention (flash-attention style).
// Block = 8 waves = 128 query rows. 128 key rows of h1 + s2 staged in LDS per
// iteration (shared by all 8 waves -> 8x less L2 traffic). Each wave keeps a
// 16x64 f32 accumulator (4 WMMA C tiles) + per-row softmax denominator.
// ---------------------------------------------------------------------------
__global__ __launch_bounds__(256) void k_attn(
    const float* __restrict__ h1, const float* __restrict__ s1,
    const float* __restrict__ s2, const float* __restrict__ s2max,
    float* __restrict__ out)
{
    __shared__ __align__(16) float sh_h1[JCHUNK * OUT_F];
    __shared__ float sh_s2[JCHUNK];

    const int tid  = threadIdx.x;
    const int lane = tid & 31;
    const int wave = tid >> 5;
    const int lm   = lane & 15;
    const int half = lane >> 4;
    const int ko   = half * 2;
    const int i0   = (blockIdx.x * 8 + wave) * 16;

    const float s1v = s1[i0 + lm];                 // query-row score, m = lm
    const float mv  = lrelu(s1v + s2max[0]);       // exact row max of f(s1+s2)

    v8f   c[4] = {};
    float dacc = 0.f;                              // partial softmax denominator

    for (int j0 = 0; j0 < N_NODES; j0 += JCHUNK) {
        __syncthreads();
        // Stage 128 key rows of h1 (32 KB) + their s2 into LDS, coalesced b128.
        const float* src = h1 + (size_t)j0 * OUT_F;
        #pragma unroll
        for (int it = 0; it < (JCHUNK * OUT_F) / (256 * 4); ++it) {
            int idx = (it * 256 + tid) * 4;
            *(v4f*)(sh_h1 + idx) = *(const v4f*)(src + idx);
        }
        if (tid < JCHUNK) sh_s2[tid] = s2[j0 + tid];
        __syncthreads();

        #pragma unroll 4
        for (int jj = 0; jj < JCHUNK; jj += 4) {
            // A fragment: e[m=lm][k=ko..ko+1] = exp(f(s1+s2) - m_i), all <= 1.
            float t0 = s1v + sh_s2[jj + ko];
            float t1 = s1v + sh_s2[jj + ko + 1];
            float e0 = __expf(lrelu(t0) - mv);
            float e1 = __expf(lrelu(t1) - mv);
            v2f a; a.x = e0; a.y = e1;
            dacc += e0 + e1;

            const float* brow = sh_h1 + (jj + ko) * OUT_F + lm;
            #pragma unroll
            for (int t = 0; t < 4; ++t) {
                v2f b;                              // B: V0 = key (jj+ko), V1 = key (jj+ko+1)
                b.x = brow[16 * t];
                b.y = brow[OUT_F + 16 * t];
                c[t] = wmma4f32(a, b, c[t]);
            }
        }
    }

    // Combine denominator halves (k-split lives in lane ^ 16), then per-row
    // broadcast: output row (r + half*8) needs denom held by lane (r + half*8).
    float d = dacc + __shfl_xor(dacc, 16, 32);     // d = denom[m = lm], all lanes
    #pragma unroll
    for (int r = 0; r < 8; ++r) {
        float rden = 1.0f / __shfl(d, r + (half << 3), 32);
        const int row = i0 + r + half * 8;
        #pragma unroll
        for (int t = 0; t < 4; ++t)
            out[(size_t)row * OUT_F + 16 * t + lm] = c[t][r] * rden;
    }
}

// ---------------------------------------------------------------------------
extern "C" void kernel_launch(void* const* d_in, const int* in_sizes, int n_in,
                              void* d_out, int out_size, void* d_ws, size_t ws_size,
                              hipStream_t stream) {
    const float* x   = (const float*)d_in[0];   // [8192, 512]
    const float* n   = (const float*)d_in[1];   // [8192, 512]
    const float* W   = (const float*)d_in[2];   // [64, 512]
    const float* wa1 = (const float*)d_in[3];   // [64]
    const float* wa2 = (const float*)d_in[4];   // [64]
    float* out = (float*)d_out;                 // [8192, 64]

    // Workspace layout (~2.07 MB): h1 | s1 | s2 | s2max
    float* h1  = (float*)d_ws;
    float* s1  = h1 + (size_t)N_NODES * OUT_F;
    float* s2  = s1 + N_NODES;
    float* s2m = s2 + N_NODES;

    dim3 blk(256);
    dim3 grd(N_NODES / 128);   // 64 blocks, 8 waves each, 16 rows per wave

    k_proj <<<grd, blk, 0, stream>>>(x, n, W, wa1, wa2, h1, s1, s2);
    k_s2max<<<1,   blk, 0, stream>>>(s2, s2m);
    k_attn <<<grd, blk, 0, stream>>>(h1, s1, s2, s2m, out);
}